// AttnWordsDecoder_1889785610887
// MI455X (gfx1250) — compile-verified
//
#include <hip/hip_runtime.h>
#include <math.h>

typedef __attribute__((ext_vector_type(2))) float v2f;
typedef __attribute__((ext_vector_type(8))) float v8f;

#define HDIM  1024
#define XDIM  2048
#define G3    3072
#define LSEQ  256
#define NSTEP 25
#define MPAD  32
#define NVOCAB 50257

__device__ __forceinline__ float waveReduceSum(float v) {
    #pragma unroll
    for (int off = 16; off > 0; off >>= 1)
        v += __shfl_xor(v, off, 32);
    return v;
}

// Build X (32x2048, padded rows zero), copy h0, zero VEC.
__global__ void init_kernel(const float* __restrict__ qv, const float* __restrict__ dec,
                            const int* __restrict__ ta, const float* __restrict__ emb,
                            float* __restrict__ X, float* __restrict__ h0,
                            float* __restrict__ VEC) {
    int gid = blockIdx.x * blockDim.x + threadIdx.x;   // 65536 threads
    int row = gid >> 11;
    int col = gid & 2047;
    float x = 0.f;
    if (row < NSTEP) {
        if (col < HDIM) {
            x = qv[col];
        } else {
            int idx = (row == 0) ? 0 : ta[row - 1];
            x = emb[(size_t)idx * HDIM + (col - HDIM)];
        }
    }
    X[gid] = x;
    if (gid < HDIM) h0[gid] = dec[gid];
    if (gid < MPAD * HDIM) VEC[gid] = 0.f;
}

// C(M x N) = A(M x K) @ W^T + bias, W row-major (N x K).
// One wave per 16x16 output tile, FP32 WMMA 16x16x4.
// Ragged-N handling: clamp the W row (garbage only lands in columns we
// never store), keeping the K-loop branch-free / full-EXEC.
// K must be a multiple of 32.
__global__ void wmma_gemm_bt(const float* __restrict__ A, int lda,
                             const float* __restrict__ W, int ldw,
                             const float* __restrict__ bias,
                             float* __restrict__ C, int ldc,
                             int Mtiles, int Ntiles, int K,
                             int Mvalid, int Nvalid) {
    int lane = threadIdx.x & 31;
    int wave = threadIdx.x >> 5;
    int tile = blockIdx.x * 8 + wave;          // blockDim.x == 256 -> 8 waves
    if (tile >= Mtiles * Ntiles) return;
    int tm = tile / Ntiles;
    int tn = tile - tm * Ntiles;
    int m0 = tm << 4, n0 = tn << 4;
    int l15   = lane & 15;
    int khalf = (lane >> 4) << 1;              // 0 or 2
    int m = m0 + l15;                          // A row for this lane
    int n = n0 + l15;                          // B col (= W row) for this lane
    int nc = (n < Nvalid) ? n : (Nvalid - 1);  // clamped: always a valid row
    const float* __restrict__ Arow = A + (size_t)m * lda + khalf;
    const float* __restrict__ Wrow = W + (size_t)nc * ldw + khalf;
    v8f acc = {};
    for (int k0 = 0; k0 < K; k0 += 32) {
        __builtin_prefetch(Wrow + k0 + 256, 0, 1);   // global_prefetch_b8, ~1KB ahead
        #pragma unroll
        for (int kk = 0; kk < 32; kk += 4) {
            v2f a, b;
            a.x = Arow[k0 + kk];
            a.y = Arow[k0 + kk + 1];
            b.x = Wrow[k0 + kk];
            b.y = Wrow[k0 + kk + 1];
            acc = __builtin_amdgcn_wmma_f32_16x16x4_f32(
                /*neg_a=*/false, a, /*neg_b=*/false, b,
                /*c_mod=*/(short)0, acc, /*reuse_a=*/false, /*reuse_b=*/false);
        }
    }
    if (n < Nvalid) {
        float bn = bias ? bias[n] : 0.f;
        int rbase = (lane >> 4) << 3;          // rows: 0..7 (lanes 0-15), 8..15 (lanes 16-31)
        #pragma unroll
        for (int v = 0; v < 8; ++v) {
            int mr = m0 + rbase + v;
            if (mr < Mvalid)
                C[(size_t)mr * ldc + n] = acc[v] + bn;
        }
    }
}

// Fused GRU step: wave i computes gh[i], gh[i+H], gh[i+2H] dots + gates.
__global__ void gru_kernel(const float* __restrict__ h, const float* __restrict__ g,
                           const float* __restrict__ Whh, const float* __restrict__ bhh,
                           float* __restrict__ hn) {
    int lane = threadIdx.x & 31;
    int i = (blockIdx.x * blockDim.x + threadIdx.x) >> 5;  // row 0..1023
    const float* __restrict__ w0 = Whh + (size_t)i * HDIM;
    const float* __restrict__ w1 = Whh + (size_t)(i + HDIM) * HDIM;
    const float* __restrict__ w2 = Whh + (size_t)(i + 2 * HDIM) * HDIM;
    float s0 = 0.f, s1 = 0.f, s2 = 0.f;
    for (int k = lane; k < HDIM; k += 32) {
        float hk = h[k];
        s0 += w0[k] * hk;
        s1 += w1[k] * hk;
        s2 += w2[k] * hk;
    }
    s0 = waveReduceSum(s0);
    s1 = waveReduceSum(s1);
    s2 = waveReduceSum(s2);
    if (lane == 0) {
        float gh_r = s0 + bhh[i];
        float gh_z = s1 + bhh[i + HDIM];
        float gh_n = s2 + bhh[i + 2 * HDIM];
        float r = 1.f / (1.f + expf(-(g[i] + gh_r)));
        float z = 1.f / (1.f + expf(-(g[i + HDIM] + gh_z)));
        float nn = tanhf(g[i + 2 * HDIM] + r * gh_n);
        hn[i] = (1.f - z) * nn + z * h[i];
    }
}

// y = W @ x + b, wave per row.
__global__ void matvec_kernel(const float* __restrict__ W, const float* __restrict__ x,
                              const float* __restrict__ b, float* __restrict__ y, int Kdim) {
    int lane = threadIdx.x & 31;
    int i = (blockIdx.x * blockDim.x + threadIdx.x) >> 5;
    const float* __restrict__ w = W + (size_t)i * Kdim;
    float s = 0.f;
    for (int k = lane; k < Kdim; k += 32) s += w[k] * x[k];
    s = waveReduceSum(s);
    if (lane == 0) y[i] = s + (b ? b[i] : 0.f);
}

// e[j] = sum_d tanh(keys[j][d] + q2[d]) * v[d]
__global__ void attn_e_kernel(const float* __restrict__ KEYS, const float* __restrict__ q2,
                              const float* __restrict__ av, float* __restrict__ e) {
    int lane = threadIdx.x & 31;
    int j = (blockIdx.x * blockDim.x + threadIdx.x) >> 5;  // 0..255
    const float* __restrict__ kr = KEYS + (size_t)j * HDIM;
    float s = 0.f;
    for (int d = lane; d < HDIM; d += 32)
        s += tanhf(kr[d] + q2[d]) * av[d];
    s = waveReduceSum(s);
    if (lane == 0) e[j] = s;
}

// softmax over 256 (one block of 256)
__global__ void softmax_kernel(const float* __restrict__ e, float* __restrict__ w) {
    __shared__ float red[256];
    int j = threadIdx.x;
    float ej = e[j];
    red[j] = ej; __syncthreads();
    for (int s = 128; s > 0; s >>= 1) {
        if (j < s) red[j] = fmaxf(red[j], red[j + s]);
        __syncthreads();
    }
    float mx = red[0]; __syncthreads();
    float ex = expf(ej - mx);
    red[j] = ex; __syncthreads();
    for (int s = 128; s > 0; s >>= 1) {
        if (j < s) red[j] += red[j + s];
        __syncthreads();
    }
    w[j] = ex / red[0];
}

// ctx[d] = sum_j w[j] * qw[j][d]
__global__ void ctx_kernel(const float* __restrict__ w, const float* __restrict__ qw,
                           float* __restrict__ ctx) {
    int d = blockIdx.x * blockDim.x + threadIdx.x;  // 0..1023
    float s = 0.f;
    for (int j = 0; j < LSEQ; ++j) s += w[j] * qw[(size_t)j * HDIM + d];
    ctx[d] = s;
}

// vec[i] = tanh(concat_W[i] . [h_new ; ctx] + cb[i])
__global__ void cvec_kernel(const float* __restrict__ hn, const float* __restrict__ ctx,
                            const float* __restrict__ CW, const float* __restrict__ cb,
                            float* __restrict__ vec_out) {
    int lane = threadIdx.x & 31;
    int i = (blockIdx.x * blockDim.x + threadIdx.x) >> 5;
    const float* __restrict__ w = CW + (size_t)i * XDIM;
    float s = 0.f;
    for (int k = lane; k < HDIM; k += 32) s += w[k] * hn[k];
    for (int k = lane; k < HDIM; k += 32) s += w[HDIM + k] * ctx[k];
    s = waveReduceSum(s);
    if (lane == 0) vec_out[i] = tanhf(s + cb[i]);
}

// in-place log_softmax over VOCAB + argmax -> out[t]; one block per step
__global__ void lsm_kernel(float* __restrict__ out) {
    __shared__ float sval[1024];
    __shared__ int   sidx[1024];
    int t = blockIdx.x;
    float* __restrict__ lg = out + NSTEP + (size_t)t * NVOCAB;
    int tid = threadIdx.x;
    float mx = -INFINITY; int mi = 0;
    for (int v = tid; v < NVOCAB; v += 1024) {
        float x = lg[v];
        if (x > mx) { mx = x; mi = v; }
    }
    sval[tid] = mx; sidx[tid] = mi; __syncthreads();
    for (int s = 512; s > 0; s >>= 1) {
        if (tid < s) {
            float ov = sval[tid + s]; int oi = sidx[tid + s];
            if (ov > sval[tid] || (ov == sval[tid] && oi < sidx[tid])) {
                sval[tid] = ov; sidx[tid] = oi;
            }
        }
        __syncthreads();
    }
    float gmx = sval[0]; int gi = sidx[0];
    __syncthreads();
    float sum = 0.f;
    for (int v = tid; v < NVOCAB; v += 1024) sum += expf(lg[v] - gmx);
    sval[tid] = sum; __syncthreads();
    for (int s = 512; s > 0; s >>= 1) {
        if (tid < s) sval[tid] += sval[tid + s];
        __syncthreads();
    }
    float lse = logf(sval[0]);
    for (int v = tid; v < NVOCAB; v += 1024) lg[v] = lg[v] - gmx - lse;
    if (tid == 0) out[t] = (float)gi;
}

extern "C" void kernel_launch(void* const* d_in, const int* in_sizes, int n_in,
                              void* d_out, int out_size, void* d_ws, size_t ws_size,
                              hipStream_t stream) {
    (void)in_sizes; (void)n_in; (void)out_size; (void)ws_size;
    const float* qw  = (const float*)d_in[0];   // (256,1,1024)
    const float* qv  = (const float*)d_in[1];   // (1,1,1024)
    const float* dec = (const float*)d_in[2];   // (1,1,1024)
    const int*   ta  = (const int*)d_in[3];     // (25,)
    const float* emb = (const float*)d_in[4];   // (50257,1024)
    const float* Wih = (const float*)d_in[5];   // (3072,2048)
    const float* Whh = (const float*)d_in[6];   // (3072,1024)
    const float* bih = (const float*)d_in[7];   // (3072,)
    const float* bhh = (const float*)d_in[8];   // (3072,)
    const float* Wq  = (const float*)d_in[9];   // (1024,1024)
    const float* bq  = (const float*)d_in[10];  // (1024,)
    const float* Wk  = (const float*)d_in[11];  // (1024,1024)
    const float* bk  = (const float*)d_in[12];  // (1024,)
    const float* av  = (const float*)d_in[13];  // (1024,)
    const float* CW  = (const float*)d_in[14];  // (1024,2048)
    const float* cb  = (const float*)d_in[15];  // (1024,)
    const float* oW  = (const float*)d_in[16];  // (50257,1024)
    const float* ob  = (const float*)d_in[17];  // (50257,)
    float* out = (float*)d_out;
    float* ws  = (float*)d_ws;

    float* X    = ws;                 // 32*2048   = 65536
    float* G    = X + 65536;          // 32*3072   = 98304
    float* KEYS = G + 98304;          // 256*1024  = 262144
    float* VEC  = KEYS + 262144;      // 32*1024   = 32768
    float* h0   = VEC + 32768;        // 1024
    float* h1   = h0 + 1024;          // 1024
    float* q2   = h1 + 1024;          // 1024
    float* e    = q2 + 1024;          // 256
    float* w    = e + 256;            // 256
    float* ctx  = w + 256;            // 1024

    // Phase 0: build inputs
    init_kernel<<<256, 256, 0, stream>>>(qv, dec, ta, emb, X, h0, VEC);

    // Phase 1: batched GEMMs (FP32 WMMA)
    // G = X @ Wih^T + bih : 2x192 tiles, K=2048
    wmma_gemm_bt<<<48, 256, 0, stream>>>(X, XDIM, Wih, XDIM, bih, G, G3,
                                         2, 192, XDIM, MPAD, G3);
    // KEYS = qw @ Wk^T + bk : 16x64 tiles, K=1024
    wmma_gemm_bt<<<128, 256, 0, stream>>>(qw, HDIM, Wk, HDIM, bk, KEYS, HDIM,
                                          16, 64, HDIM, LSEQ, HDIM);

    // Phase 2: sequential recurrence (L2-resident matvecs)
    float* hb[2] = {h0, h1};
    for (int t = 0; t < NSTEP; ++t) {
        const float* hin = hb[t & 1];
        float* hout = hb[(t + 1) & 1];
        gru_kernel<<<128, 256, 0, stream>>>(hin, G + (size_t)t * G3, Whh, bhh, hout);
        matvec_kernel<<<128, 256, 0, stream>>>(Wq, hout, bq, q2, HDIM);
        attn_e_kernel<<<32, 256, 0, stream>>>(KEYS, q2, av, e);
        softmax_kernel<<<1, 256, 0, stream>>>(e, w);
        ctx_kernel<<<4, 256, 0, stream>>>(w, qw, ctx);
        cvec_kernel<<<128, 256, 0, stream>>>(hout, ctx, CW, cb, VEC + (size_t)t * HDIM);
    }

    // Phase 3: batched logits GEMM, out_W streamed exactly once (~206 MB)
    // LOGITS(25 x 50257) written directly into d_out at offset NSTEP.
    // tiles: 2 x ceil(50257/16)=3142 -> 6284 tiles -> 786 blocks of 8 waves
    wmma_gemm_bt<<<786, 256, 0, stream>>>(VEC, HDIM, oW, HDIM, ob, out + NSTEP, NVOCAB,
                                          2, 3142, HDIM, NSTEP, NVOCAB);

    // Phase 4: in-place log_softmax + argmax
    lsm_kernel<<<NSTEP, 1024, 0, stream>>>(out);
}